// MResConvPoint_12790412607771
// MI455X (gfx1250) — compile-verified
//
#include <hip/hip_runtime.h>

#define BB    2
#define CH    64
#define NV    150000
#define KNB   8
#define KF    192          // 3 * 64 feature length
#define NSTEP 48           // KF / 4
#define VPB   128          // vertices per block
#define NBLK  ((NV + VPB - 1) / VPB)

typedef __attribute__((ext_vector_type(2))) float v2f;
typedef __attribute__((ext_vector_type(4))) float v4f;
typedef __attribute__((ext_vector_type(8))) float v8f;

// ---------------------------------------------------------------------------
// Kernel 1: transpose x [B, C, N] -> xT [B, N, C]   (C = 64)
// ---------------------------------------------------------------------------
__global__ void transpose_kernel(const float* __restrict__ x, float* __restrict__ xT) {
  __shared__ float tile[64][65];
  const int nblk = (NV + 63) / 64;
  const int b  = blockIdx.x / nblk;
  const int n0 = (blockIdx.x % nblk) * 64;
  const int tid = threadIdx.x;
#pragma unroll
  for (int i = 0; i < 16; i++) {
    int idx = i * 256 + tid;
    int c = idx >> 6, n = idx & 63;
    if (n0 + n < NV) tile[n][c] = x[(long)(b * CH + c) * NV + n0 + n];
  }
  __syncthreads();
#pragma unroll
  for (int i = 0; i < 16; i++) {
    int idx = i * 256 + tid;
    int n = idx >> 6, c = idx & 63;
    if (n0 + n < NV) xT[(long)(b * NV + n0 + n) * CH + c] = tile[n][c];
  }
}

// ---------------------------------------------------------------------------
// Kernel 2: swizzle weights into per-lane B-fragment layout, zero BN sums.
// B frag (f32 16x16x4): lane<16 holds K={k0,k0+1}, lane>=16 holds K={k0+2,k0+3}
// at column N = tile*16 + lane%16.  Stored as float2 per (step, tile, lane).
// Wt[k][o] = W[o][c][kw], k = kw*64 + c.
// ---------------------------------------------------------------------------
__global__ void prep_kernel(const float* __restrict__ W0, const float* __restrict__ W1,
                            float* __restrict__ Bsw0, float* __restrict__ Bsw1,
                            float* __restrict__ gred) {
  const int tid = threadIdx.x;
  if (tid < 128) gred[tid] = 0.0f;
  for (int idx = tid; idx < 2 * NSTEP * 4 * 32; idx += 256) {
    const int conv = idx / (NSTEP * 4 * 32);
    const int e    = idx - conv * (NSTEP * 4 * 32);
    const int s    = e >> 7;          // / 128
    const int t    = (e >> 5) & 3;
    const int lane = e & 31;
    const float* W = conv ? W1 : W0;
    float* Bsw     = conv ? Bsw1 : Bsw0;
    const int o  = t * 16 + (lane & 15);
    const int k0 = s * 4 + ((lane >> 4) << 1);
    const int k1 = k0 + 1;
    Bsw[e * 2 + 0] = W[(o * 64 + (k0 & 63)) * 3 + (k0 >> 6)];
    Bsw[e * 2 + 1] = W[(o * 64 + (k1 & 63)) * 3 + (k1 >> 6)];
  }
}

// ---------------------------------------------------------------------------
// Kernel 3/5: mesh conv via V_WMMA_F32_16X16X4_F32.
//   SECOND=false: src=xT, write y0T, accumulate relu stats into gred.
//   SECOND=true : src=y0T with z=relu(v)*scale+shift applied on load,
//                 residual-add y0T, relu, write out [B, C, N].
// ---------------------------------------------------------------------------
template <bool SECOND>
__global__ __launch_bounds__(256)
void conv_kernel(const float* __restrict__ src,   // [B, N, 64]
                 const int*   __restrict__ nbrs,  // [B, N, 8]
                 const float* __restrict__ Bsw,   // swizzled weights (12288 floats)
                 const float* __restrict__ scale, // [64] (SECOND)
                 const float* __restrict__ shift, // [64] (SECOND)
                 float* __restrict__ y0T,         // out (pass0) / residual in (pass1)
                 float* __restrict__ out,         // [B, C, N] final (pass1)
                 float* __restrict__ gred)        // [128] global sums (pass0)
{
  __shared__ float Afrag[8 * NSTEP * 32 * 2];  // 96 KB: per-wave A fragments
  __shared__ float Bls[NSTEP * 4 * 32 * 2];    // 48 KB: B fragments
  __shared__ float red[128];                   // BN partial sums

  const int tid = threadIdx.x;
  const int b   = blockIdx.x / NBLK;
  const int v0  = (blockIdx.x % NBLK) * VPB;
  const bool full = (v0 + VPB) <= NV;          // uniform: no bounds checks needed

  if (tid < 128) red[tid] = 0.0f;

  { // stage weights to LDS (coalesced b128)
    const v4f* s4 = (const v4f*)Bsw;
    v4f* d4 = (v4f*)Bls;
#pragma unroll
    for (int i = 0; i < 12; i++) d4[i * 256 + tid] = s4[i * 256 + tid];
  }

  // ---------------- phase 1: build features into A-fragment layout ---------
  {
    const int vloc = tid >> 1;          // 0..127
    const int half = tid & 1;           // which 32 channels
    const int vg   = v0 + vloc;
    const bool ok  = vg < NV;
    const int w    = vloc >> 4;         // wave that consumes this vertex
    const int m    = vloc & 15;         // row within 16x16 tile
    int nb[KNB];
#pragma unroll
    for (int j = 0; j < KNB; j++) nb[j] = 0;
    if (ok) {
      const int* np = nbrs + (long)(b * NV + vg) * KNB;
#pragma unroll
      for (int j = 0; j < KNB; j++) nb[j] = np[j];
    }
    const float* crow = src + (long)(b * NV + vg) * CH;

#pragma unroll
    for (int ch4 = 0; ch4 < 4; ch4++) {
      const int c0 = half * 32 + ch4 * 8;
      float ctr[8], ns[8], ad[8], scv[8], shv[8];
      if (SECOND) {
        const v4f* sp = (const v4f*)(scale + c0);
        const v4f* hp = (const v4f*)(shift + c0);
        v4f s0 = sp[0], s1 = sp[1], h0 = hp[0], h1 = hp[1];
#pragma unroll
        for (int i = 0; i < 4; i++) { scv[i] = s0[i]; scv[4 + i] = s1[i];
                                      shv[i] = h0[i]; shv[4 + i] = h1[i]; }
      }
#pragma unroll
      for (int i = 0; i < 8; i++) { ctr[i] = 0.0f; ns[i] = 0.0f; ad[i] = 0.0f; }
      if (ok) {
        { const v4f* cp = (const v4f*)(crow + c0);
          v4f a = cp[0], c = cp[1];
#pragma unroll
          for (int i = 0; i < 4; i++) { ctr[i] = a[i]; ctr[4 + i] = c[i]; } }
        if (SECOND) {
#pragma unroll
          for (int i = 0; i < 8; i++)
            ctr[i] = __builtin_fmaf(fmaxf(ctr[i], 0.0f), scv[i], shv[i]);
        }
#pragma unroll
        for (int j = 0; j < KNB; j++) {
          const v4f* npv = (const v4f*)(src + (long)(b * NV + nb[j]) * CH + c0);
          v4f qa = npv[0], qb = npv[1];
          float q[8];
#pragma unroll
          for (int i = 0; i < 4; i++) { q[i] = qa[i]; q[4 + i] = qb[i]; }
#pragma unroll
          for (int i = 0; i < 8; i++) {
            float qq = SECOND ? __builtin_fmaf(fmaxf(q[i], 0.0f), scv[i], shv[i]) : q[i];
            ns[i] += qq;
            ad[i] += fabsf(ctr[i] - qq);
          }
        }
      }
      // scatter into swizzled A-fragment LDS
#pragma unroll
      for (int i = 0; i < 8; i++) {
        const int c = c0 + i;
#pragma unroll
        for (int kw = 0; kw < 3; kw++) {
          const float val = (kw == 0) ? ctr[i] : ((kw == 1) ? ns[i] : ad[i]);
          const int k    = kw * 64 + c;
          const int s    = k >> 2;
          const int rr   = k & 3;
          const int lane = m + ((rr & 2) << 3);   // +16 for K upper pair
          const int r    = rr & 1;
          Afrag[(((w * NSTEP) + s) * 32 + lane) * 2 + r] = val;
        }
      }
    }
  }
  __syncthreads();

  // ---------------- phase 2: WMMA GEMM (16 verts x 64 ch per wave) ---------
  const int w    = tid >> 5;
  const int lane = tid & 31;
  const v2f* Af = ((const v2f*)Afrag) + (w * NSTEP) * 32 + lane;
  const v2f* Bf = ((const v2f*)Bls) + lane;
  v8f acc[4];
#pragma unroll
  for (int t = 0; t < 4; t++) acc[t] = (v8f)(0.0f);

  for (int s = 0; s < NSTEP; s++) {
    const v2f a = Af[s * 32];                       // conflict-free ds_load_b64
#pragma unroll
    for (int t = 0; t < 4; t++) {
      const v2f bf = Bf[(s * 4 + t) * 32];
      acc[t] = __builtin_amdgcn_wmma_f32_16x16x4_f32(
          false, a, false, bf, (short)0, acc[t], false, false);
    }
  }

  // ---------------- phase 3: epilogue ------------------------------------
  {
    // D layout: lane holds channel ch = t*16 + lane%16, vertices vbase..vbase+7
    const int vbase = v0 + w * 16 + ((lane >> 4) << 3);
#pragma unroll
    for (int t = 0; t < 4; t++) {
      const int ch = t * 16 + (lane & 15);
      if (!SECOND) {
        // BN stats: unconditional (padded vertices contribute exactly 0)
        float psum = 0.0f, psq = 0.0f;
#pragma unroll
        for (int r = 0; r < 8; r++) {
          const float rv = fmaxf(acc[t][r], 0.0f);
          psum += rv; psq += rv * rv;
        }
        psum += __shfl_xor(psum, 16, 32);        // fold lane halves (same ch)
        psq  += __shfl_xor(psq, 16, 32);
        if (lane < 16) { atomicAdd(&red[ch], psum); atomicAdd(&red[64 + ch], psq); }
        // store y0T [B, N, 64] (stride-64 b32, 64B-coalesced across half-wave)
        float* yp = y0T + (long)(b * NV + vbase) * CH + ch;
        if (full) {
#pragma unroll
          for (int r = 0; r < 8; r++) yp[r * CH] = acc[t][r];
        } else {
#pragma unroll
          for (int r = 0; r < 8; r++)
            if (vbase + r < NV) yp[r * CH] = acc[t][r];
        }
      } else {
        // residual + relu; out [B, C, N]: lane's 8 values are CONTIGUOUS in N
        const float* rp = y0T + (long)(b * NV + vbase) * CH + ch;
        float* op = out + (long)(b * CH + ch) * NV + vbase;
        if (full) {
          v4f o0, o1;
#pragma unroll
          for (int r = 0; r < 4; r++)
            o0[r] = fmaxf(acc[t][r] + rp[r * CH], 0.0f);
#pragma unroll
          for (int r = 0; r < 4; r++)
            o1[r] = fmaxf(acc[t][4 + r] + rp[(4 + r) * CH], 0.0f);
          __builtin_nontemporal_store(o0, (v4f*)op);      // 2x global_store_b128
          __builtin_nontemporal_store(o1, (v4f*)op + 1);
        } else {
#pragma unroll
          for (int r = 0; r < 8; r++)
            if (vbase + r < NV) op[r] = fmaxf(acc[t][r] + rp[r * CH], 0.0f);
        }
      }
    }
  }
  if (!SECOND) {
    __syncthreads();
    if (tid < 128) atomicAdd(&gred[tid], red[tid]);
  }
}

// ---------------------------------------------------------------------------
// Kernel 4: finalize BN: scale = gamma*rsqrt(var+eps), shift = beta - mean*scale
// ---------------------------------------------------------------------------
__global__ void bnfin_kernel(const float* __restrict__ gred,
                             const float* __restrict__ gamma,
                             const float* __restrict__ beta,
                             float* __restrict__ scale, float* __restrict__ shift) {
  const int c = threadIdx.x;
  if (c < 64) {
    const float M    = (float)(BB * NV);
    const float mean = gred[c] / M;
    const float var  = gred[64 + c] / M - mean * mean;   // biased variance
    const float sc   = gamma[c] * rsqrtf(var + 1e-5f);
    scale[c] = sc;
    shift[c] = beta[c] - mean * sc;
  }
}

// ---------------------------------------------------------------------------
extern "C" void kernel_launch(void* const* d_in, const int* in_sizes, int n_in,
                              void* d_out, int out_size, void* d_ws, size_t ws_size,
                              hipStream_t stream) {
  const float* x     = (const float*)d_in[0];
  const int*   nbrs  = (const int*)d_in[1];
  const float* W0    = (const float*)d_in[2];
  const float* W1    = (const float*)d_in[3];
  const float* gamma = (const float*)d_in[4];
  const float* beta  = (const float*)d_in[5];
  float* out = (float*)d_out;

  float* ws    = (float*)d_ws;
  float* xT    = ws;                                // B*N*64
  float* y0T   = xT  + (size_t)BB * NV * CH;        // B*N*64
  float* Bsw0  = y0T + (size_t)BB * NV * CH;        // 12288
  float* Bsw1  = Bsw0 + NSTEP * 4 * 32 * 2;         // 12288
  float* gred  = Bsw1 + NSTEP * 4 * 32 * 2;         // 128
  float* scale = gred + 128;                        // 64
  float* shf   = scale + 64;                        // 64

  const int nblk64 = (NV + 63) / 64;
  transpose_kernel<<<BB * nblk64, 256, 0, stream>>>(x, xT);
  prep_kernel<<<1, 256, 0, stream>>>(W0, W1, Bsw0, Bsw1, gred);
  conv_kernel<false><<<BB * NBLK, 256, 0, stream>>>(xT, nbrs, Bsw0, nullptr, nullptr,
                                                    y0T, nullptr, gred);
  bnfin_kernel<<<1, 64, 0, stream>>>(gred, gamma, beta, scale, shf);
  conv_kernel<true><<<BB * NBLK, 256, 0, stream>>>(y0T, nbrs, Bsw1, scale, shf,
                                                   y0T, out, nullptr);
}